// GINModel_24867860644187
// MI455X (gfx1250) — compile-verified
//
#include <hip/hip_runtime.h>

// ---------------------------------------------------------------------------
// GIN edge scorer for MI455X (gfx1250, wave32).
//
//   1. aggr[d,:] += x[s,:] over 800K edges  (HBM/L2-atomic bound stage)
//   2. h1 = relu(((1+eps)*x + aggr) @ W1 + b1)   -- fp32 WMMA 16x16x4
//   3. h2 = relu(h1 @ W2 + b2)                   -- fp32 WMMA 16x16x4, A-strip
//                                                   staged via TENSOR_LOAD_TO_LDS
//   4. nodeA[i] = h2[i,:].fcW[:256]; nodeB[i] = h2[i,:].fcW[256:]
//   5. out[e] = nodeA[src[e]] + nodeB[dst[e]] + fcb    (factored edge MLP)
//
// GEMM blocking: 256 threads (8 waves) own 48 rows x 256 cols. Each wave:
// 2 col-tiles x 3 row-tiles -> every B-fragment load feeds 3 WMMAs, cutting
// W L2 traffic from 800MB to ~267MB per GEMM. Only the last block is partial
// (32 rows); its third row-tile computes on don't-care LDS data and the
// stores are guarded, so EXEC stays all-1s for every WMMA.
//
// Workspace: 2 * 50000*256*4 = 102,400,000 bytes.
// ---------------------------------------------------------------------------

#define N_NODES 50000
#define N_FEAT  256
#define HIDDEN  256
#define N_EDGES 800000

#define ROWS_PER_BLOCK 48   // 3 row-tiles of 16
#define LDS_STRIDE     260  // 256 + 4 pad floats -> spreads the 64 LDS banks

typedef __attribute__((ext_vector_type(2))) float v2f;
typedef __attribute__((ext_vector_type(8))) float v8f;
typedef __attribute__((ext_vector_type(4))) unsigned int v4u;
typedef __attribute__((ext_vector_type(4))) int v4i;
typedef __attribute__((ext_vector_type(8))) int v8i;

#if __has_builtin(__builtin_amdgcn_tensor_load_to_lds) && \
    __has_builtin(__builtin_amdgcn_s_wait_tensorcnt)
#define GIN_TDM 1
#endif

// ---------------------------------------------------------------------------
// Stage 1: scatter-add aggregation. 64 threads per edge, float4 per thread.
// aggr (51MB) is L2-resident (192MB L2) so the fp32 atomics resolve on-chip.
// ---------------------------------------------------------------------------
__global__ __launch_bounds__(256)
void gin_scatter(const float4* __restrict__ x4, const int* __restrict__ ei,
                 float* __restrict__ aggr)
{
    const long long t = (long long)blockIdx.x * 256 + threadIdx.x;
    const int e  = (int)(t >> 6);
    const int f4 = (int)(t & 63);
    if (e >= N_EDGES) return;
    const int s = ei[e];
    const int d = ei[N_EDGES + e];
    const float4 v = x4[(size_t)s * 64 + f4];
    float* dst = aggr + (size_t)d * 256 + f4 * 4;
    atomicAdd(dst + 0, v.x);
    atomicAdd(dst + 1, v.y);
    atomicAdd(dst + 2, v.z);
    atomicAdd(dst + 3, v.w);
}

// ---------------------------------------------------------------------------
// Stages 2 & 3: OUT = relu(A @ W + bias), A optionally fused as aggr+(1+eps)x.
// WMMA f32 16x16x4 fragment layouts (ISA 7.12.2), lane = 16*h + m:
//   A vgpr j = A[m][k + 2h + j]          (16x4, row striped in lane)
//   B vgpr j = W[k + 2h + j][col0 + m]   (4x16, row striped across lanes)
//   D vgpr v = OUT[row0 + v + 8h][col0 + m]
// When the input needs no fusion, the 48x256 A-strip (incl. the 4-DWORD row
// pad) is DMA'd into LDS by the Tensor Data Mover: pad_interval=7 (pad every
// 256 DWORDs), pad_amount=3 (4 DWORDs) reproduces the 260-float row stride.
// ---------------------------------------------------------------------------
__global__ __launch_bounds__(256)
void gin_gemm_relu(const float* __restrict__ Ain,
                   const float* __restrict__ Xfuse,   // nullptr => plain A
                   const float* __restrict__ epsPtr,  // only read when Xfuse
                   const float* __restrict__ W,
                   const float* __restrict__ bias,
                   float* __restrict__ Out)
{
    __shared__ float As[ROWS_PER_BLOCK * LDS_STRIDE];
    const int row0  = blockIdx.x * ROWS_PER_BLOCK;
    const int tid   = threadIdx.x;
    const int nrows = (N_NODES - row0 < ROWS_PER_BLOCK) ? (N_NODES - row0)
                                                        : ROWS_PER_BLOCK;

    bool staged = false;
#ifdef GIN_TDM
    if (!Xfuse) {
        if ((tid >> 5) == 0) {   // one wave issues the DMA (uniform branch;
                                 // tensor ops ignore EXEC, so gate by wave)
            const unsigned lds = (unsigned)(unsigned long long)(&As[0]);
            const unsigned long long ga =
                (unsigned long long)(const void*)(Ain + (size_t)row0 * 256);
            // D# group0: count=1 | lds_addr | global_addr(57b) | type=2
            v4u g0 = { 1u, lds, (unsigned)ga,
                       (unsigned)((ga >> 32) & 0x01FFFFFFull) | 0x80000000u };
            // D# group1: data_size=4B, pad_enable, pad_interval=7, pad_amount=3
            //            tensor_dim0=256, tensor_dim1=nrows,
            //            tile_dim0=256, tile_dim1=nrows, dim0_stride=256
            v8i g1 = { 0x07D20000,
                       0x01000000,       // tensor_dim0 = 256  (bits 79:48)
                       nrows << 16,      // tensor_dim1        (bits 111:80)
                       0x01000000,       // tile_dim0 = 256    (bits 127:112)
                       nrows,            // tile_dim1          (bits 143:128)
                       256, 0, 0 };      // tensor_dim0_stride (bits 207:160)
            v4i gz = { 0, 0, 0, 0 };
#if __has_include(<hip/amd_detail/amd_gfx1250_TDM.h>)
            v8i g5 = { 0, 0, 0, 0, 0, 0, 0, 0 };
            __builtin_amdgcn_tensor_load_to_lds(g0, g1, gz, gz, g5, 0);
#else
            __builtin_amdgcn_tensor_load_to_lds(g0, g1, gz, gz, 0);
#endif
            __builtin_amdgcn_s_wait_tensorcnt(0);
        }
        staged = true;
    }
#endif
    if (!staged) {
        float scale = 0.0f;
        if (Xfuse) scale = 1.0f + epsPtr[0];
        for (int i = tid; i < nrows * 256; i += 256) {
            const int r = i >> 8, c = i & 255;
            const size_t g = (size_t)(row0 + r) * 256 + c;
            float v = Ain[g];
            if (Xfuse) v += scale * Xfuse[g];
            As[r * LDS_STRIDE + c] = v;
        }
    }
    __syncthreads();

    const int wave = tid >> 5;
    const int lane = tid & 31;
    const int h = lane >> 4;         // lane half selects K pair / row half
    const int m = lane & 15;
    const int colBase = wave * 32;   // each wave: 2 column tiles of 16

    v8f acc[2][3] = {};

    for (int k = 0; k < 256; k += 4) {
        const int kk = k + 2 * h;
        v2f a[3];
#pragma unroll
        for (int rt = 0; rt < 3; ++rt) {
            a[rt].x = As[(rt * 16 + m) * LDS_STRIDE + kk];
            a[rt].y = As[(rt * 16 + m) * LDS_STRIDE + kk + 1];
        }
#pragma unroll
        for (int t = 0; t < 2; ++t) {
            const int col = colBase + t * 16 + m;
            v2f b;
            b.x = W[(size_t)kk * 256 + col];
            b.y = W[(size_t)(kk + 1) * 256 + col];
#pragma unroll
            for (int rt = 0; rt < 3; ++rt)
                acc[t][rt] = __builtin_amdgcn_wmma_f32_16x16x4_f32(
                    false, a[rt], false, b, (short)0, acc[t][rt], false, false);
        }
    }

#pragma unroll
    for (int t = 0; t < 2; ++t) {
        const int col = colBase + t * 16 + m;
        const float bv = bias[col];
#pragma unroll
        for (int rt = 0; rt < 3; ++rt) {
#pragma unroll
            for (int v = 0; v < 8; ++v) {
                const int row = row0 + rt * 16 + v + 8 * h;
                if (row < N_NODES) {   // guard only stores (tail block)
                    float val = acc[t][rt][v] + bv;
                    Out[(size_t)row * 256 + col] = val > 0.0f ? val : 0.0f;
                }
            }
        }
    }
}

// ---------------------------------------------------------------------------
// Stage 4: per-node dual dot with fcW halves. One wave per node; each lane
// handles 8 contiguous features (two float4 loads), then shuffle-reduce.
// ---------------------------------------------------------------------------
__global__ __launch_bounds__(256)
void gin_nodedot(const float* __restrict__ h2, const float* __restrict__ fcW,
                 float* __restrict__ nodeA, float* __restrict__ nodeB)
{
    const int node = blockIdx.x * 8 + (threadIdx.x >> 5);
    const int lane = threadIdx.x & 31;
    if (node >= N_NODES) return;

    const float4* row = (const float4*)(h2 + (size_t)node * 256);
    const float4 u = row[lane * 2];
    const float4 w = row[lane * 2 + 1];
    const int c = lane * 8;

    const float* fa = fcW;        // src half
    const float* fb = fcW + 256;  // dst half
    float sa = u.x * fa[c]     + u.y * fa[c + 1] + u.z * fa[c + 2] + u.w * fa[c + 3]
             + w.x * fa[c + 4] + w.y * fa[c + 5] + w.z * fa[c + 6] + w.w * fa[c + 7];
    float sb = u.x * fb[c]     + u.y * fb[c + 1] + u.z * fb[c + 2] + u.w * fb[c + 3]
             + w.x * fb[c + 4] + w.y * fb[c + 5] + w.z * fb[c + 6] + w.w * fb[c + 7];

    for (int off = 16; off > 0; off >>= 1) {
        sa += __shfl_down(sa, off);
        sb += __shfl_down(sb, off);
    }
    if (lane == 0) {
        nodeA[node] = sa;
        nodeB[node] = sb;
    }
}

// ---------------------------------------------------------------------------
// Stage 5: per-edge scalar gather (6.4MB instead of a 3.3GB gather-GEMM).
// ---------------------------------------------------------------------------
__global__ __launch_bounds__(256)
void gin_edge_score(const int* __restrict__ ei,
                    const float* __restrict__ nodeA,
                    const float* __restrict__ nodeB,
                    const float* __restrict__ fcb,
                    float* __restrict__ out)
{
    const int e = blockIdx.x * 256 + threadIdx.x;
    if (e >= N_EDGES) return;
    out[e] = nodeA[ei[e]] + nodeB[ei[N_EDGES + e]] + fcb[0];
}

// ---------------------------------------------------------------------------
extern "C" void kernel_launch(void* const* d_in, const int* in_sizes, int n_in,
                              void* d_out, int out_size, void* d_ws, size_t ws_size,
                              hipStream_t stream)
{
    const float* x   = (const float*)d_in[0];
    const int*   ei  = (const int*)  d_in[1];
    const float* W1  = (const float*)d_in[2];
    const float* b1  = (const float*)d_in[3];
    const float* W2  = (const float*)d_in[4];
    const float* b2  = (const float*)d_in[5];
    const float* fcW = (const float*)d_in[6];
    const float* fcb = (const float*)d_in[7];
    const float* eps = (const float*)d_in[8];
    float* out = (float*)d_out;

    const size_t stripBytes = (size_t)N_NODES * HIDDEN * sizeof(float); // 51.2 MB
    char* ws = (char*)d_ws;
    float* aggr  = (float*)ws;                 // stage 1 output / GEMM1 input
    float* h1    = (float*)(ws + stripBytes);  // GEMM1 output
    float* h2    = aggr;                       // GEMM2 output (aggr is dead)
    float* nodeA = h1;                         // stage 4 output (h1 is dead)
    float* nodeB = h1 + N_NODES;

    const int gemmBlocks = (N_NODES + ROWS_PER_BLOCK - 1) / ROWS_PER_BLOCK; // 1042

    // 1) aggregation
    hipMemsetAsync(aggr, 0, stripBytes, stream);
    {
        const long long threads = (long long)N_EDGES * 64;
        const int blocks = (int)((threads + 255) / 256);   // 200000
        gin_scatter<<<blocks, 256, 0, stream>>>((const float4*)x, ei, aggr);
    }
    // 2) h1 = relu(((1+eps)x + aggr) @ W1 + b1)   (fused input path)
    gin_gemm_relu<<<gemmBlocks, 256, 0, stream>>>(aggr, x, eps, W1, b1, h1);
    // 3) h2 = relu(h1 @ W2 + b2)                  (TDM-staged A strip)
    gin_gemm_relu<<<gemmBlocks, 256, 0, stream>>>(h1, nullptr, nullptr, W2, b2, h2);
    // 4) per-node dots
    gin_nodedot<<<(N_NODES + 7) / 8, 256, 0, stream>>>(h2, fcW, nodeA, nodeB);
    // 5) per-edge scores
    gin_edge_score<<<(N_EDGES + 255) / 256, 256, 0, stream>>>(ei, nodeA, nodeB, fcb, out);
}